// WaveletTransformLayer_44805098831996
// MI455X (gfx1250) — compile-verified
//
#include <hip/hip_runtime.h>
#include <math.h>

// db4 decomposition filters (rec filters are these un-reversed, per reference)
constexpr float DEC_LO[8] = {-0.010597401784997278f,  0.032883011666982945f,
                              0.030841381835986965f, -0.18703481171888114f,
                             -0.02798376941698385f,   0.6308807679295904f,
                              0.7148465705525415f,    0.23037781330885523f};
constexpr float DEC_HI[8] = {-0.23037781330885523f,   0.7148465705525415f,
                             -0.6308807679295904f,   -0.02798376941698385f,
                              0.18703481171888114f,   0.030841381835986965f,
                             -0.032883011666982945f, -0.010597401784997278f};

#define TLEN 4096
#define LA   2051      // number of cA/cD coefficients per series
#define NSER 8192      // 256 batches * 32 channels
#define CH1  129       // phase-1 chunk (16*129 = 2064 >= 2051)
#define QCH  128       // phase-3 chunk: 128 q-steps -> 256 outputs

// e (symmetric half-sample extension, minus first sample) -> x time index.
// valid e indices: 0..4108 ; clamps above (clamped values are never used).
__device__ __forceinline__ int eidx(int i) {
  i = i > 4108 ? 4108 : i;
  if (i < 6)    return 5 - i;          // left reflection
  if (i > 4101) return 8197 - i;       // right reflection (2T+5-i)
  return i - 6;                        // interior
}

// ---------------- Phase 1: per-(series,chunk) partial sums of cD, cD^2 -----
__global__ void k_dwt_stats(const float* __restrict__ in,
                            float* __restrict__ ps, float* __restrict__ pq) {
  int tid   = blockIdx.x * blockDim.x + threadIdx.x;
  int c     = tid & 31;
  int b     = (tid >> 5) & 255;
  int chunk = tid >> 13;                       // 0..15
  const float* xp = in + (size_t)b * (TLEN * 32) + c;   // lane=channel: coalesced

  int k0 = chunk * CH1;
  int k1 = k0 + CH1; if (k1 > LA) k1 = LA;

  float w[8];
#pragma unroll
  for (int j = 0; j < 8; ++j) w[j] = xp[eidx(2 * k0 + j) * 32];

  float s = 0.f, q = 0.f;
  for (int k = k0; k < k1; ++k) {
    float cd = 0.f;
#pragma unroll
    for (int j = 0; j < 8; ++j) cd = fmaf(w[j], DEC_HI[7 - j], cd);
    s += cd; q = fmaf(cd, cd, q);

    int i = 2 * k + 8;
    int pf = i + 96;                           // prefetch ~48 iters ahead
    if (pf <= 4101) __builtin_prefetch(xp + (pf - 6) * 32, 0, 3);
#pragma unroll
    for (int j = 0; j < 6; ++j) w[j] = w[j + 2];
    w[6] = xp[eidx(i) * 32];
    w[7] = xp[eidx(i + 1) * 32];
  }
  int series = (b << 5) | c;
  ps[series * 16 + chunk] = s;
  pq[series * 16 + chunk] = q;
}

// ---------------- Phase 2: WMMA reduction of 16 partials -> threshold ------
typedef __attribute__((ext_vector_type(2))) float v2f;
typedef __attribute__((ext_vector_type(8))) float v8f;

__global__ void k_reduce_thr(const float* __restrict__ ps,
                             const float* __restrict__ pq,
                             float* __restrict__ thr) {
  int tid  = blockIdx.x * blockDim.x + threadIdx.x;
  int wave = tid >> 5;                         // 512 waves, 16 series each
  int lane = threadIdx.x & 31;
  int S0   = wave * 16;
  int m    = lane & 15;                        // A-matrix row (M)
  int kp   = (lane >> 4) << 1;                 // lanes<16 hold K{0,1}, else K{2,3}

  v2f ones; ones.x = 1.f; ones.y = 1.f;        // B = all-ones 4x16 -> row sums
  v8f accS = {0.f,0.f,0.f,0.f,0.f,0.f,0.f,0.f};
  v8f accQ = {0.f,0.f,0.f,0.f,0.f,0.f,0.f,0.f};

#pragma unroll
  for (int cc = 0; cc < 4; ++cc) {             // chain K=4 four times -> 16
    int kk = cc * 4 + kp;
    const float* pA = ps + (S0 + m) * 16 + kk;
    v2f A; A.x = pA[0]; A.y = pA[1];
    accS = __builtin_amdgcn_wmma_f32_16x16x4_f32(false, A, false, ones,
                                                 (short)0, accS, false, false);
    const float* pB = pq + (S0 + m) * 16 + kk;
    v2f Aq; Aq.x = pB[0]; Aq.y = pB[1];
    accQ = __builtin_amdgcn_wmma_f32_16x16x4_f32(false, Aq, false, ones,
                                                 (short)0, accQ, false, false);
  }

  // D layout: VGPR r, lane 0 -> M=r (N=0); lane 16 -> M=r+8 (N=0)
  const float invN = 1.0f / 2051.0f;
  if (lane == 0 || lane == 16) {
    int base = S0 + ((lane >> 4) << 3);
#pragma unroll
    for (int i = 0; i < 8; ++i) {
      float ssum = accS[i], qsum = accQ[i];
      float mean = ssum * invN;
      float var  = fmaxf(qsum * invN - mean * mean, 0.f);
      thr[base + i] = 2.0f * sqrtf(var);       // 2 * population std
    }
  }
}

// ---------------- Phase 3: recompute coeffs, soft-threshold, idwt ----------
__global__ void k_recon(const float* __restrict__ in,
                        const float* __restrict__ thr,
                        float* __restrict__ out) {
  int tid   = blockIdx.x * blockDim.x + threadIdx.x;
  int c     = tid & 31;
  int b     = (tid >> 5) & 255;
  int chunk = tid >> 13;
  const float* xp = in  + (size_t)b * (TLEN * 32) + c;
  float*       op = out + (size_t)b * (TLEN * 32) + c;
  float t = thr[(b << 5) | c];

  int q0 = chunk * QCH;
  float w[8];
#pragma unroll
  for (int j = 0; j < 8; ++j) w[j] = xp[eidx(2 * q0 + j) * 32];

  float a[4], d[4];
#pragma unroll
  for (int u = 0; u < 3; ++u) {                // warm up cA/cD[q0..q0+2]
    float ca = 0.f, cd = 0.f;
#pragma unroll
    for (int j = 0; j < 8; ++j) {
      ca = fmaf(w[j], DEC_LO[7 - j], ca);
      cd = fmaf(w[j], DEC_HI[7 - j], cd);
    }
    a[u] = ca;
    d[u] = copysignf(fmaxf(fabsf(cd) - t, 0.f), cd);
    int i = 2 * (q0 + u) + 8;
#pragma unroll
    for (int j = 0; j < 6; ++j) w[j] = w[j + 2];
    w[6] = xp[eidx(i) * 32];
    w[7] = xp[eidx(i + 1) * 32];
  }

  for (int q = q0; q < q0 + QCH; ++q) {
    float ca = 0.f, cd = 0.f;                  // k = q+3 from e[2q+6 .. 2q+13]
#pragma unroll
    for (int j = 0; j < 8; ++j) {
      ca = fmaf(w[j], DEC_LO[7 - j], ca);
      cd = fmaf(w[j], DEC_HI[7 - j], cd);
    }
    a[3] = ca;
    d[3] = copysignf(fmaxf(fabsf(cd) - t, 0.f), cd);

    float y0 = 0.f, y1 = 0.f;                  // y[2q] (odd taps), y[2q+1] (even taps)
#pragma unroll
    for (int dk = 0; dk < 4; ++dk) {
      y0 = fmaf(DEC_LO[2 * dk + 1], a[dk], y0);
      y0 = fmaf(DEC_HI[2 * dk + 1], d[dk], y0);
      y1 = fmaf(DEC_LO[2 * dk], a[dk], y1);
      y1 = fmaf(DEC_HI[2 * dk], d[dk], y1);
    }
    __builtin_nontemporal_store(y0, op + (2 * q) * 32);      // keep input in L2
    __builtin_nontemporal_store(y1, op + (2 * q + 1) * 32);

    a[0] = a[1]; a[1] = a[2]; a[2] = a[3];
    d[0] = d[1]; d[1] = d[2]; d[2] = d[3];
    int i = 2 * q + 14;
    int pf = i + 96;
    if (pf <= 4101) __builtin_prefetch(xp + (pf - 6) * 32, 0, 3);
#pragma unroll
    for (int j = 0; j < 6; ++j) w[j] = w[j + 2];
    w[6] = xp[eidx(i) * 32];
    w[7] = xp[eidx(i + 1) * 32];
  }
}

extern "C" void kernel_launch(void* const* d_in, const int* in_sizes, int n_in,
                              void* d_out, int out_size, void* d_ws, size_t ws_size,
                              hipStream_t stream) {
  const float* in = (const float*)d_in[0];
  float* out = (float*)d_out;
  float* ws  = (float*)d_ws;
  float* ps  = ws;                    // 131072 floats: partial sums of cD
  float* pq  = ws + NSER * 16;        // 131072 floats: partial sums of cD^2
  float* thr = ws + 2 * NSER * 16;    // 8192 floats: per-series thresholds

  // 8192 series * 16 chunks = 131072 threads per streaming phase
  k_dwt_stats <<<512, 256, 0, stream>>>(in, ps, pq);
  k_reduce_thr<<< 64, 256, 0, stream>>>(ps, pq, thr);   // 512 full waves, WMMA
  k_recon     <<<512, 256, 0, stream>>>(in, thr, out);
}